// KalmanFilter_32667521253783
// MI455X (gfx1250) — compile-verified
//
#include <hip/hip_runtime.h>

typedef float v8f __attribute__((ext_vector_type(8)));
typedef float v2f __attribute__((ext_vector_type(2)));

#define TT 512
#define NB 2048

static constexpr long MEANS_OFF = 0;
static constexpr long COVS_OFF  = (long)NB * TT * 8;                 // 8388608
static constexpr long RS_OFF    = COVS_OFF + (long)NB * TT * 64;     // 75497472
static constexpr long HS_OFF    = RS_OFF + (long)NB * TT * 4;        // 79691776

__device__ __forceinline__ float rfl(float x) {
  return __int_as_float(__builtin_amdgcn_readfirstlane(__float_as_int(x)));
}

// One wave handles 4 groups: their 8x8 covariances are the 4 blocks of a
// 16x16 tile held in WMMA C-layout (lane = tile column, VGPRs = rows).
// Predict step: cov_p = F_bd * cov_u * F_bd^T via two chained
// V_WMMA_F32_16X16X4_F32 K-slices (A = blockdiag(F,F), constant).
__global__ __launch_bounds__(256) void kf_kernel(
    const float* __restrict__ xin,   // [B,T,2]
    const float* __restrict__ mean0, // [B,8]
    const float* __restrict__ cov0,  // [B,8,8]
    const float* __restrict__ Fm,    // [8,8]
    const float* __restrict__ Hm,    // [2,8]
    const float* __restrict__ Qm,    // [8,8]
    const float* __restrict__ Rm,    // [2,2]
    float* __restrict__ out)
{
  __shared__ float tbuf[8][32 * 33];   // per-wave 16x16 f32 tile, stride-33 padded
  const int lane = threadIdx.x & 31;
  const int wid  = threadIdx.x >> 5;
  const int hi   = lane >> 4;          // tile row-half (block row i)
  const int l16  = lane & 15;          // tile column
  const int jblk = l16 >> 3;           // block col j
  const int c    = lane & 7;           // column within the 8x8 block
  const int g    = hi * 2 + jblk;      // group index within wave
  const long b   = ((long)blockIdx.x * 8 + wid) * 4 + g;

  // Wave-uniform constants pinned to SGPRs
  float Fs[64];
#pragma unroll
  for (int i = 0; i < 64; ++i) Fs[i] = rfl(Fm[i]);
  float Hsc[16];
#pragma unroll
  for (int i = 0; i < 16; ++i) Hsc[i] = rfl(Hm[i]);
  const float R00 = rfl(Rm[0]), R01 = rfl(Rm[1]), R11 = rfl(Rm[3]);

  // Per-lane constants: Q column c, F row c
  float qcol[8], fr[8];
#pragma unroll
  for (int d = 0; d < 8; ++d) { qcol[d] = Qm[d * 8 + c]; fr[d] = Fm[c * 8 + d]; }

  // A operand slices of F_bd = blockdiag(F,F), per documented 16x4 f32 A-layout:
  // lane m (mod 16), slice j: VGPR0 holds K=4j(+2 for hi half), VGPR1 holds K+1.
  float a0[4], a1[4];
#pragma unroll
  for (int j = 0; j < 4; ++j) {
    const int m = l16;
    const int k0 = 4 * j + 2 * hi;
    const int k1 = k0 + 1;
    a0[j] = ((m >> 3) == (k0 >> 3)) ? Fm[(m & 7) * 8 + (k0 & 7)] : 0.0f;
    a1[j] = ((m >> 3) == (k1 >> 3)) ? Fm[(m & 7) * 8 + (k1 & 7)] : 0.0f;
  }

  // State: P = column c of this group's cov (== C-layout tile slot), replicated mean
  float P[8], mv[8];
#pragma unroll
  for (int d = 0; d < 8; ++d) P[d] = cov0[b * 64 + d * 8 + c];
#pragma unroll
  for (int d = 0; d < 8; ++d) mv[d] = mean0[b * 8 + d];
  float scalarM = mean0[b * 8 + c];

  const int gb4 = (lane & 24) * 4;     // bpermute byte base of this group's lane block

  const float* xp     = xin + b * TT * 2;
  float* mout         = out + MEANS_OFF + b * TT * 8 + c;
  float* covout       = out + COVS_OFF + b * TT * 64 + (long)c * 8;

#pragma unroll 1
  for (int t = 0; t < TT; ++t) {
    // ---- emit 1-step-ahead prediction (prior state) ----
    mout[(long)t * 8] = scalarM;
    // cov symmetric: this lane's column c == row c -> contiguous 32B
    *(float4*)(covout + (long)t * 64)     = make_float4(P[0], P[1], P[2], P[3]);
    *(float4*)(covout + (long)t * 64 + 4) = make_float4(P[4], P[5], P[6], P[7]);

    const float2 xv = *(const float2*)(xp + t * 2);

    // ---- measurement update ----
    float hp0 = 0.f, hp1 = 0.f;                // HP[m, c]
#pragma unroll
    for (int d = 0; d < 8; ++d) { hp0 += Hsc[d] * P[d]; hp1 += Hsc[8 + d] * P[d]; }

    float hpv0[8], hpv1[8];                    // HP rows, gathered across group lanes
#pragma unroll
    for (int d = 0; d < 8; ++d) {
      hpv0[d] = __int_as_float(__builtin_amdgcn_ds_bpermute(gb4 + 4 * d, __float_as_int(hp0)));
      hpv1[d] = __int_as_float(__builtin_amdgcn_ds_bpermute(gb4 + 4 * d, __float_as_int(hp1)));
    }

    float s00 = R00, s01 = R01, s11 = R11;     // S = HP H^T + R (2x2)
#pragma unroll
    for (int d = 0; d < 8; ++d) {
      s00 += hpv0[d] * Hsc[d];
      s01 += hpv0[d] * Hsc[8 + d];
      s11 += hpv1[d] * Hsc[8 + d];
    }
    const float det  = s00 * s11 - s01 * s01;
    const float rdet = 1.0f / det;             // IEEE divide (precision)
    const float i00 = s11 * rdet, i01 = -s01 * rdet, i11 = s00 * rdet;

    float q0 = 0.f, q1 = 0.f;                  // H * mean
#pragma unroll
    for (int d = 0; d < 8; ++d) { q0 += Hsc[d] * mv[d]; q1 += Hsc[8 + d] * mv[d]; }
    const float r0 = xv.x - q0, r1 = xv.y - q1;

#pragma unroll
    for (int d = 0; d < 8; ++d) {              // Kt rows; rank-2 cov downdate; mean update
      const float k0v = i00 * hpv0[d] + i01 * hpv1[d];
      const float k1v = i01 * hpv0[d] + i11 * hpv1[d];
      P[d]  -= k0v * hp0 + k1v * hp1;          // cov_u column c
      mv[d] += k0v * r0 + k1v * r1;            // mean_u (replicated)
    }

    // ---- mean predict: mean_p = F * mean_u (SGPR F x replicated vector) ----
    float mn[8];
#pragma unroll
    for (int d = 0; d < 8; ++d) {
      float a = 0.f;
#pragma unroll
      for (int e = 0; e < 8; ++e) a += Fs[d * 8 + e] * mv[e];
      mn[d] = a;
    }
    float sm = 0.f;
#pragma unroll
    for (int e = 0; e < 8; ++e) sm += fr[e] * mv[e];
    scalarM = sm;
#pragma unroll
    for (int d = 0; d < 8; ++d) mv[d] = mn[d];

    // ---- cov predict via WMMA ----
    // C-layout -> B-layout of the same tile: half-lane swap + select.
    float sx[8];
#pragma unroll
    for (int v = 0; v < 8; ++v) sx[v] = __shfl_xor(P[v], 16, 32);
    float bb[8];
    bb[0] = hi ? sx[2] : P[0];  bb[1] = hi ? sx[3] : P[1];
    bb[2] = hi ? sx[6] : P[4];  bb[3] = hi ? sx[7] : P[5];
    bb[4] = hi ? P[2] : sx[0];  bb[5] = hi ? P[3] : sx[1];
    bb[6] = hi ? P[6] : sx[4];  bb[7] = hi ? P[7] : sx[5];

    v8f acc = {0.f, 0.f, 0.f, 0.f, 0.f, 0.f, 0.f, 0.f};   // X1 = F_bd * cov_u
#pragma unroll
    for (int j = 0; j < 4; ++j) {
      v2f A; A.x = a0[j]; A.y = a1[j];
      v2f Bv; Bv.x = bb[2 * j]; Bv.y = bb[2 * j + 1];
      acc = __builtin_amdgcn_wmma_f32_16x16x4_f32(false, A, false, Bv,
                                                  (short)0, acc, false, false);
    }

    // Per-block transpose of X1 through LDS (store C-layout, gather B-layout of Z,
    // Z_ij = (X1_ij)^T). Same-wave DS ops are in-order; no barrier needed.
#pragma unroll
    for (int v = 0; v < 8; ++v) tbuf[wid][lane * 33 + v] = acc[v];

    float zz[8];
#pragma unroll
    for (int v = 0; v < 8; ++v) {
      const int k  = 4 * (v >> 1) + (v & 1) + 2 * hi;   // B-layout k for this reg
      const int i2 = k >> 3, kl = k & 7;
      zz[v] = tbuf[wid][(8 * jblk + kl + 16 * i2) * 33 + c];
    }

    v8f acc2 = {0.f, 0.f, 0.f, 0.f, 0.f, 0.f, 0.f, 0.f}; // F_bd * Z = F P F^T per block
#pragma unroll
    for (int j = 0; j < 4; ++j) {
      v2f A; A.x = a0[j]; A.y = a1[j];
      v2f Bv; Bv.x = zz[2 * j]; Bv.y = zz[2 * j + 1];
      acc2 = __builtin_amdgcn_wmma_f32_16x16x4_f32(false, A, false, Bv,
                                                   (short)0, acc2, false, false);
    }
#pragma unroll
    for (int d = 0; d < 8; ++d) P[d] = acc2[d] + qcol[d]; // + Q
  }
}

// Rs/Hs are pure broadcasts: bandwidth-only fill kernel.
__global__ __launch_bounds__(256) void fill_rh(const float* __restrict__ Hm,
                                               const float* __restrict__ Rm,
                                               float* __restrict__ out)
{
  const long tid = (long)blockIdx.x * blockDim.x + threadIdx.x;   // over B*T
  if (tid >= (long)NB * TT) return;
  const float4 rv = *(const float4*)Rm;
  ((float4*)(out + RS_OFF))[tid] = rv;
  const float4* h4 = (const float4*)Hm;
  float4* o = (float4*)(out + HS_OFF) + tid * 4;
  o[0] = h4[0]; o[1] = h4[1]; o[2] = h4[2]; o[3] = h4[3];
}

extern "C" void kernel_launch(void* const* d_in, const int* in_sizes, int n_in,
                              void* d_out, int out_size, void* d_ws, size_t ws_size,
                              hipStream_t stream) {
  (void)in_sizes; (void)n_in; (void)out_size; (void)d_ws; (void)ws_size;
  const float* xin   = (const float*)d_in[0];
  const float* mean0 = (const float*)d_in[1];
  const float* cov0  = (const float*)d_in[2];
  const float* Fm    = (const float*)d_in[3];
  const float* Hm    = (const float*)d_in[4];
  const float* Qm    = (const float*)d_in[5];
  const float* Rm    = (const float*)d_in[6];
  float* out = (float*)d_out;

  // 2048 groups / 4 per wave = 512 waves = 64 blocks x 256 threads
  kf_kernel<<<64, 256, 0, stream>>>(xin, mean0, cov0, Fm, Hm, Qm, Rm, out);
  fill_rh<<<(NB * TT + 255) / 256, 256, 0, stream>>>(Hm, Rm, out);
}